// Indexer_76630806495676
// MI455X (gfx1250) — compile-verified
//
#include <hip/hip_runtime.h>

// Problem constants
#define B_   2
#define S_   1024
#define D_   7168
#define H_   64
#define HD_  128
#define RD_  64
#define QL_  1536
#define T_   (B_ * S_)        // 2048
#define NQ_  (H_ * HD_)       // 8192
#define EPS_ 1e-6f
#define SCALE_ 0.08838834764831845f  // 128^-0.5

typedef float v2f __attribute__((ext_vector_type(2)));
typedef float v8f __attribute__((ext_vector_type(8)));
typedef unsigned int u32x4 __attribute__((ext_vector_type(4)));
typedef int          i32x8 __attribute__((ext_vector_type(8)));
typedef int          i32x4 __attribute__((ext_vector_type(4)));

#define AS3 __attribute__((address_space(3)))

// ---- CDNA5 feature detection ----------------------------------------------
#if defined(__has_builtin)
#  if __has_builtin(__builtin_amdgcn_global_load_async_to_lds_b128)
#    define HAVE_ASYNC_LDS 1
#  endif
#  if __has_builtin(__builtin_amdgcn_tensor_load_to_lds) && \
      __has_builtin(__builtin_amdgcn_s_wait_tensorcnt)
#    define HAVE_TDM 1
#  endif
#endif

#if defined(__has_builtin) && __has_builtin(__builtin_amdgcn_s_wait_asynccnt)
#  define WAIT_ASYNC0() __builtin_amdgcn_s_wait_asynccnt(0)
#else
#  define WAIT_ASYNC0() asm volatile("s_wait_asynccnt 0x0" ::: "memory")
#endif

static __device__ __forceinline__ v8f wmma_f32(v2f a, v2f b, v8f c) {
  // D = A(16x4,f32) * B(4x16,f32) + C(16x16,f32)
  return __builtin_amdgcn_wmma_f32_16x16x4_f32(false, a, false, b, (short)0, c,
                                               false, false);
}

#if defined(HAVE_ASYNC_LDS)
static __device__ __forceinline__ void async_cp16(const float* g, float* l) {
  // builtin takes generic int4* pointers (global src, LDS dst), imm offset,
  // imm cpol
  __builtin_amdgcn_global_load_async_to_lds_b128((i32x4*)g, (i32x4*)l, 0, 0);
}
#endif

#if defined(HAVE_TDM)
// TDM 2D tile load: tile_w x tile_h f32 elements, row stride `stride_elems`
// in global memory, into LDS at byte offset lds_off with +4 dwords of LDS
// padding appended per 128-dword row (pad_interval=6 -> 128 dw,
// pad_amount=3 -> 4 dw)  => LDS row stride 132 floats.
static __device__ __forceinline__ void tdm_load_2d(const float* gaddr,
                                                   unsigned lds_off,
                                                   unsigned tile_w,
                                                   unsigned tile_h,
                                                   unsigned stride_elems) {
  unsigned long long ga = (unsigned long long)(uintptr_t)gaddr;
  u32x4 g0;
  g0[0] = 1u;                                        // count=1, user D#
  g0[1] = lds_off;                                   // lds_addr (bytes)
  g0[2] = (unsigned)(ga & 0xFFFFFFFFu);              // global_addr[31:0]
  g0[3] = (unsigned)((ga >> 32) & 0x01FFFFFFu)       // global_addr[56:32]
        | (2u << 30);                                // type=2 ("image")
  i32x8 g1;
  g1[0] = (int)((2u << 16)                           // data_size = 4 bytes
              | (1u << 20)                           // pad_enable
              | (6u << 22)                           // pad_interval: 128 dw
              | (3u << 25));                         // pad_amount: 4 dw
  g1[1] = (int)(tile_w << 16);                       // tensor_dim0[15:0]
  g1[2] = (int)(tile_h << 16);                       // tensor_dim1[15:0]
  g1[3] = (int)(tile_w << 16);                       // tile_dim0
  g1[4] = (int)(tile_h & 0xFFFFu);                   // tile_dim1 (tile_dim2=0)
  g1[5] = (int)stride_elems;                         // tensor_dim0_stride[31:0]
  g1[6] = 0;
  g1[7] = 0;
  i32x4 z4 = {0, 0, 0, 0};
  i32x8 z8 = {0, 0, 0, 0, 0, 0, 0, 0};
  // 6-arg form: (g0, g1, g2, g3, g?, cpol)
  __builtin_amdgcn_tensor_load_to_lds(g0, g1, z4, z4, z8, 0);
}
#endif

// ---------------------------------------------------------------------------
// Kernel A: q = qr[T,QL] @ wq[QL,NQ], fused interleaved RoPE on the first
// RD dims of each head. 8 waves x (16x16 WMMA tile) = 16x128 block tile.
// A-tile double-buffered in LDS via async global->LDS copies.
// ---------------------------------------------------------------------------
__global__ void __launch_bounds__(256)
k_gemm_q_rope(const float* __restrict__ qr, const float* __restrict__ wq,
              const float* __restrict__ cosb, const float* __restrict__ sinb,
              float* __restrict__ qout)
{
  __shared__ float a_lds[2][16][68];
  const int tid  = threadIdx.x;
  const int lane = tid & 31;
  const int wave = tid >> 5;
  const int t0   = blockIdx.y * 16;
  const int n0   = blockIdx.x * 128 + wave * 16;
  const int nn   = lane & 15;
  const int half = lane >> 4;
  const int kq   = half * 2;

  const int lrow = tid >> 4;                       // 0..15
  const int lk4  = (tid & 15) * 4;                 // 0..60
  const float* asrc = qr + (size_t)(t0 + lrow) * QL_ + lk4;

#if defined(HAVE_ASYNC_LDS)
  async_cp16(asrc, &a_lds[0][lrow][lk4]);          // preload chunk 0
  WAIT_ASYNC0();
  __syncthreads();
#endif

  v8f c = {};
  int buf = 0;
  for (int kc = 0; kc < QL_; kc += 64, buf ^= 1) {
#if defined(HAVE_ASYNC_LDS)
    if (kc + 64 < QL_)                             // prefetch next chunk
      async_cp16(asrc + kc + 64, &a_lds[buf ^ 1][lrow][lk4]);
#else
    __syncthreads();
    *(float4*)&a_lds[0][lrow][lk4] = *(const float4*)(asrc + kc);
    __syncthreads();
#endif
    const float* bp = wq + (size_t)kc * NQ_ + n0 + nn;
    #pragma unroll 4
    for (int kk = 0; kk < 64; kk += 4) {
      v2f a, b;
      a.x = a_lds[buf][nn][kk + kq];
      a.y = a_lds[buf][nn][kk + kq + 1];
      b.x = bp[(size_t)(kk + kq) * NQ_];
      b.y = bp[(size_t)(kk + kq + 1) * NQ_];
      c = wmma_f32(a, b, c);
    }
#if defined(HAVE_ASYNC_LDS)
    WAIT_ASYNC0();                                 // prefetch landed
    __syncthreads();
#else
    buf = 1;                                       // keep single buffer (0)
#endif
  }

  // Epilogue: interleaved RoPE (pairs live in adjacent lanes of C layout).
  const int  col = n0 + nn;
  const int  hc  = col & (HD_ - 1);
  const bool rot = (hc < RD_);
  const float sgn = (col & 1) ? 1.0f : -1.0f;
  for (int r = 0; r < 8; ++r) {
    int   m = r + half * 8;
    int   t = t0 + m;
    float v = c[r];
    float p = __shfl_xor(v, 1, 32);
    if (rot) {
      float cs = cosb[t * RD_ + hc];
      float sn = sinb[t * RD_ + hc];
      v = v * cs + sgn * p * sn;
    }
    qout[(size_t)t * NQ_ + col] = v;
  }
}

// ---------------------------------------------------------------------------
// Kernel B: [kp | w] = x[T,D] @ [wk (D,128) | wproj (D,64)]. 12 waves.
// ---------------------------------------------------------------------------
__global__ void __launch_bounds__(384)
k_gemm_kw(const float* __restrict__ x, const float* __restrict__ wk,
          const float* __restrict__ wproj, float* __restrict__ kout,
          float* __restrict__ wout)
{
  __shared__ float a_lds[2][16][68];
  const int tid  = threadIdx.x;
  const int lane = tid & 31;
  const int wave = tid >> 5;                       // 0..11
  const int t0   = blockIdx.x * 16;
  const int n0   = wave * 16;
  const int nn   = lane & 15;
  const int half = lane >> 4;
  const int kq   = half * 2;

  const bool isK = (n0 < HD_);
  const float* bp0 = isK ? wk : wproj;
  const int ldb  = isK ? HD_ : H_;
  const int bcol = isK ? (n0 + nn) : (n0 - HD_ + nn);

  const int lrow = tid >> 4;
  const int lk4  = (tid & 15) * 4;
  const float* asrc = x + (size_t)(t0 + lrow) * D_ + lk4;
  const bool loader = (tid < 256);

#if defined(HAVE_ASYNC_LDS)
  if (loader) async_cp16(asrc, &a_lds[0][lrow][lk4]);
  WAIT_ASYNC0();
  __syncthreads();
#endif

  v8f c = {};
  int buf = 0;
  for (int kc = 0; kc < D_; kc += 64, buf ^= 1) {
#if defined(HAVE_ASYNC_LDS)
    if (loader && kc + 64 < D_)
      async_cp16(asrc + kc + 64, &a_lds[buf ^ 1][lrow][lk4]);
#else
    __syncthreads();
    if (loader) *(float4*)&a_lds[0][lrow][lk4] = *(const float4*)(asrc + kc);
    __syncthreads();
#endif
    const float* bp = bp0 + (size_t)kc * ldb + bcol;
    #pragma unroll 4
    for (int kk = 0; kk < 64; kk += 4) {
      v2f a, b;
      a.x = a_lds[buf][nn][kk + kq];
      a.y = a_lds[buf][nn][kk + kq + 1];
      b.x = bp[(size_t)(kk + kq) * ldb];
      b.y = bp[(size_t)(kk + kq + 1) * ldb];
      c = wmma_f32(a, b, c);
    }
#if defined(HAVE_ASYNC_LDS)
    WAIT_ASYNC0();
    __syncthreads();
#else
    buf = 1;
#endif
  }

  for (int r = 0; r < 8; ++r) {
    int m = r + half * 8;
    int t = t0 + m;
    if (isK) kout[(size_t)t * HD_ + bcol] = c[r];
    else     wout[(size_t)t * H_  + bcol] = c[r];
  }
}

// ---------------------------------------------------------------------------
// Kernel C: per-row LayerNorm over 128 + interleaved RoPE on first 64,
// in place. One wave per row.
// ---------------------------------------------------------------------------
__global__ void __launch_bounds__(256)
k_ln_rope(float* __restrict__ kbuf, const float* __restrict__ cosb,
          const float* __restrict__ sinb, const float* __restrict__ gamma,
          const float* __restrict__ beta)
{
  const int tid  = threadIdx.x;
  const int lane = tid & 31;
  const int wave = tid >> 5;
  const int t    = blockIdx.x * 8 + wave;
  const int col  = lane * 4;

  float4 v = *(const float4*)(kbuf + (size_t)t * HD_ + col);
  float s = v.x + v.y + v.z + v.w;
  float q = v.x * v.x + v.y * v.y + v.z * v.z + v.w * v.w;
  for (int off = 16; off >= 1; off >>= 1) {
    s += __shfl_xor(s, off, 32);
    q += __shfl_xor(q, off, 32);
  }
  float mu   = s * (1.0f / HD_);
  float var  = q * (1.0f / HD_) - mu * mu;
  float rstd = rsqrtf(var + EPS_);

  float4 g  = *(const float4*)(gamma + col);
  float4 bt = *(const float4*)(beta + col);
  float4 n;
  n.x = (v.x - mu) * rstd * g.x + bt.x;
  n.y = (v.y - mu) * rstd * g.y + bt.y;
  n.z = (v.z - mu) * rstd * g.z + bt.z;
  n.w = (v.w - mu) * rstd * g.w + bt.w;

  if (col < RD_) {
    float c0 = cosb[t * RD_ + col],     s0 = sinb[t * RD_ + col];
    float c2 = cosb[t * RD_ + col + 2], s2 = sinb[t * RD_ + col + 2];
    float4 o;
    o.x = n.x * c0 - n.y * s0;
    o.y = n.y * c0 + n.x * s0;
    o.z = n.z * c2 - n.w * s2;
    o.w = n.w * c2 + n.z * s2;
    n = o;
  }
  *(float4*)(kbuf + (size_t)t * HD_ + col) = n;
}

// ---------------------------------------------------------------------------
// Kernel D: scores[b,q,k] = SCALE * sum_h w[q,h] * relu(q[q,h,:]·k[k,:]).
// Block = (b, 16-row q tile), 4 waves each own a 16-col k tile.
// K block (64x128) and per-head q tiles (16x128, row stride 8192) staged
// through LDS by the Tensor Data Mover (hardware LDS padding -> stride 132),
// q tiles double-buffered so head h+1's DMA overlaps head h's WMMAs.
// ---------------------------------------------------------------------------
__global__ void __launch_bounds__(128)
k_scores(const float* __restrict__ qg, const float* __restrict__ kg,
         const float* __restrict__ wg, float* __restrict__ out)
{
  __shared__ float klds[64][132];
  __shared__ float qlds[2][16][132];
  const int tid  = threadIdx.x;
  const int lane = tid & 31;
  const int wave = tid >> 5;                      // 0..3
  const int b    = blockIdx.x >> 6;
  const int qt   = blockIdx.x & 63;
  const int q0   = qt * 16;
  const int nn   = lane & 15;
  const int half = lane >> 4;
  const int kq   = half * 2;

  const int krow  = tid >> 1;                     // 0..63 (fallback loader)
  const int khalf = (tid & 1) * 64;
  const int qrow  = tid >> 3;                     // 0..15 (fallback loader)
  const int qseg  = (tid & 7) * 16;

#if defined(HAVE_TDM)
  const unsigned klds_off  = (unsigned)(uintptr_t)(AS3 void*)&klds[0][0];
  const unsigned qlds_off0 = (unsigned)(uintptr_t)(AS3 void*)&qlds[0][0][0];
  const unsigned qlds_off1 = (unsigned)(uintptr_t)(AS3 void*)&qlds[1][0][0];
#endif
  const float* qbase = qg + (size_t)(b * S_ + q0) * NQ_;

  for (int it = 0; it < 16; ++it) {
#if defined(HAVE_TDM)
    if (wave == 0) {
      // K block: 64 rows x 128 cols, contiguous rows (stride 128)
      tdm_load_2d(kg + (size_t)(b * S_ + it * 64) * HD_, klds_off, HD_, 64, HD_);
      // q tile head 0: 16 rows x 128 cols, row stride NQ_
      tdm_load_2d(qbase + 0 * HD_, qlds_off0, HD_, 16, NQ_);
      __builtin_amdgcn_s_wait_tensorcnt(0);
    }
    __syncthreads();
#else
    {
      const float* src = kg + (size_t)(b * S_ + it * 64 + krow) * HD_ + khalf;
      #pragma unroll
      for (int j = 0; j < 64; j += 4)
        *(float4*)&klds[krow][khalf + j] = *(const float4*)(src + j);
    }
    __syncthreads();
#endif

    v8f acc = {};
    for (int h = 0; h < H_; ++h) {
      const int buf = h & 1;
#if defined(HAVE_TDM)
      if (wave == 0 && h + 1 < H_)                 // prefetch next head's tile
        tdm_load_2d(qbase + (h + 1) * HD_, buf ? qlds_off0 : qlds_off1,
                    HD_, 16, NQ_);
#else
      const float* qs = qbase + (size_t)qrow * NQ_ + h * HD_ + qseg;
      *(float4*)&qlds[0][qrow][qseg + 0]  = *(const float4*)(qs + 0);
      *(float4*)&qlds[0][qrow][qseg + 4]  = *(const float4*)(qs + 4);
      *(float4*)&qlds[0][qrow][qseg + 8]  = *(const float4*)(qs + 8);
      *(float4*)&qlds[0][qrow][qseg + 12] = *(const float4*)(qs + 12);
      __syncthreads();
#endif

      v8f c = {};
      #pragma unroll 8
      for (int kk = 0; kk < HD_; kk += 4) {
        v2f a, bb;
#if defined(HAVE_TDM)
        a.x  = qlds[buf][nn][kk + kq];
        a.y  = qlds[buf][nn][kk + kq + 1];
#else
        a.x  = qlds[0][nn][kk + kq];
        a.y  = qlds[0][nn][kk + kq + 1];
#endif
        bb.x = klds[wave * 16 + nn][kk + kq];      // B[d][n] = k[n][d]
        bb.y = klds[wave * 16 + nn][kk + kq + 1];
        c = wmma_f32(a, bb, c);
      }

#if defined(HAVE_TDM)
      if (wave == 0) __builtin_amdgcn_s_wait_tensorcnt(0);
      __syncthreads();
#else
      __syncthreads();
#endif

      for (int r = 0; r < 8; ++r) {
        int   m = r + half * 8;
        float wgt = wg[(size_t)(b * S_ + q0 + m) * H_ + h];
        float l = c[r];
        acc[r] += wgt * (l > 0.0f ? l : 0.0f);
      }
    }

    const int ktile = it * 4 + wave;
    for (int r = 0; r < 8; ++r) {
      int m = r + half * 8;
      out[(size_t)b * S_ * S_ + (size_t)(q0 + m) * S_ + ktile * 16 + nn] =
          SCALE_ * acc[r];
    }
    __syncthreads();
  }
}

// ---------------------------------------------------------------------------
extern "C" void kernel_launch(void* const* d_in, const int* in_sizes, int n_in,
                              void* d_out, int out_size, void* d_ws, size_t ws_size,
                              hipStream_t stream) {
  const float* x     = (const float*)d_in[0];
  const float* qr    = (const float*)d_in[1];
  const float* cosb  = (const float*)d_in[2];
  const float* sinb  = (const float*)d_in[3];
  const float* wq    = (const float*)d_in[4];
  const float* wk    = (const float*)d_in[5];
  const float* wproj = (const float*)d_in[6];
  const float* gamma = (const float*)d_in[7];
  const float* beta  = (const float*)d_in[8];
  float* out = (float*)d_out;

  float* q_ws = (float*)d_ws;                      // T*NQ   (64 MB)
  float* k_ws = q_ws + (size_t)T_ * NQ_;           // T*HD   (1 MB)
  float* w_ws = k_ws + (size_t)T_ * HD_;           // T*H    (0.5 MB)

  k_gemm_q_rope<<<dim3(NQ_ / 128, T_ / 16), 256, 0, stream>>>(qr, wq, cosb, sinb, q_ws);
  k_gemm_kw   <<<dim3(T_ / 16),             384, 0, stream>>>(x, wk, wproj, k_ws, w_ws);
  k_ln_rope   <<<dim3(T_ / 8),              256, 0, stream>>>(k_ws, cosb, sinb, gamma, beta);
  k_scores    <<<dim3(B_ * (S_ / 16)),      128, 0, stream>>>(q_ws, k_ws, w_ws, out);
}